// GraphResDecoderBlock_15925738734398
// MI455X (gfx1250) — compile-verified
//
#include <hip/hip_runtime.h>

typedef float v2f __attribute__((ext_vector_type(2)));
typedef float v8f __attribute__((ext_vector_type(8)));

static constexpr int N_NODES = 50000;
static constexpr int N_EDGES = 800000;

static inline int cdiv(int a, int b) { return (a + b - 1) / b; }

__device__ __forceinline__ void fadd_agent(float* p, float v) {
  // no-return relaxed agent-scope fp32 atomic -> global_atomic_add_f32
  (void)__hip_atomic_fetch_add(p, v, __ATOMIC_RELAXED, __HIP_MEMORY_SCOPE_AGENT);
}

// ---------------- elementwise / setup kernels ----------------

__global__ void k_zero(float* __restrict__ p, int n) {
  int i = blockIdx.x * blockDim.x + threadIdx.x;
  if (i < n) p[i] = 0.f;
}

__global__ void k_copy(const float* __restrict__ a, float* __restrict__ b, int n) {
  int i = blockIdx.x * blockDim.x + threadIdx.x;
  if (i < n) b[i] = a[i];
}

__global__ void k_deg(const int* __restrict__ src, float* __restrict__ deg, int e) {
  int i = blockIdx.x * blockDim.x + threadIdx.x;
  if (i < e) fadd_agent(&deg[src[i]], 1.0f);
}

__global__ void k_wedge(const int* __restrict__ src, const int* __restrict__ dst,
                        const float* __restrict__ deg, float* __restrict__ w, int e) {
  int i = blockIdx.x * blockDim.x + threadIdx.x;
  if (i >= e) return;
  float ds = deg[src[i]], dd = deg[dst[i]];
  float a = (ds > 0.f) ? rsqrtf(ds) : 0.f;
  float b = (dd > 0.f) ? rsqrtf(dd) : 0.f;
  w[i] = -a * b;
}

// Fill T row = [t0 (fin valid, finp padded) | zeros for t1,t2 segments]
__global__ void k_fill_t0(const float* __restrict__ x, float* __restrict__ T,
                          int fin, int kpad, int total) {
  int i = blockIdx.x * blockDim.x + threadIdx.x;
  if (i >= total) return;
  int n = i / kpad, f = i - n * kpad;
  T[i] = (f < fin) ? x[(size_t)n * fin + f] : 0.f;
}

// t0 = concat(procH [N,96], enc [N,32], prev [N,3]); rest zero (kpad = 396, finp = 132)
__global__ void k_fill_t0_cat(const float* __restrict__ a, const float* __restrict__ b,
                              const float* __restrict__ c, float* __restrict__ T,
                              int kpad, int total) {
  int i = blockIdx.x * blockDim.x + threadIdx.x;
  if (i >= total) return;
  int n = i / kpad, f = i - n * kpad;
  float v = 0.f;
  if (f < 96)       v = a[(size_t)n * 96 + f];
  else if (f < 128) v = b[(size_t)n * 32 + (f - 96)];
  else if (f < 131) v = c[(size_t)n * 3 + (f - 128)];
  T[i] = v;
}

// Vectorized scatter: T[dst, offOut+f..f+3] += scale*w[e]*T[src, offIn+f..f+3]
// offIn/offOut are multiples of finp (mult of 4 floats) -> 16B-aligned float4 gathers.
__global__ void k_spmm4(const int* __restrict__ src, const int* __restrict__ dst,
                        const float* __restrict__ w, float* __restrict__ T,
                        int fin4, int kpad, int offIn, int offOut, float scale, int total) {
  int i = blockIdx.x * blockDim.x + threadIdx.x;
  if (i >= total) return;
  int e = i / fin4, g = i - e * fin4;
  int f = g << 2;
  float wv = scale * w[e];
  const float4 v = *(const float4*)(T + (size_t)src[e] * kpad + offIn + f);
  float* d = T + (size_t)dst[e] * kpad + offOut + f;
  fadd_agent(&d[0], wv * v.x);
  fadd_agent(&d[1], wv * v.y);
  fadd_agent(&d[2], wv * v.z);
  fadd_agent(&d[3], wv * v.w);
}

// t2 -= t0 (over padded width finp; pad cols are 0-0=0)
__global__ void k_t2fix(float* __restrict__ T, int finp, int kpad, int total) {
  int i = blockIdx.x * blockDim.x + threadIdx.x;
  if (i >= total) return;
  int n = i / finp, f = i - n * finp;
  T[(size_t)n * kpad + 2 * finp + f] -= T[(size_t)n * kpad + f];
}

// Wpad[kpad, fpad] <- W[3, fin, fo] with per-segment K padding and column padding
__global__ void k_packw(const float* __restrict__ W, float* __restrict__ Wp,
                        int fin, int finp, int fo, int fpad, int kpad) {
  int i = blockIdx.x * blockDim.x + threadIdx.x;
  int total = kpad * fpad;
  if (i >= total) return;
  int krow = i / fpad, n = i - krow * fpad;
  int k = krow / finp, f = krow - k * finp;
  Wp[i] = (f < fin && n < fo) ? W[((size_t)k * fin + f) * fo + n] : 0.f;
}

// ---------------- WMMA fp32 GEMM: out = A[M,K] @ B[K,Fpad] + bias (+add) ----------------
// One wave computes one 16x16 output tile via V_WMMA_F32_16X16X4_F32.
__global__ __launch_bounds__(256)
void k_gemm_wmma(const float* __restrict__ A, int lda,
                 const float* __restrict__ B, int ldb,
                 const float* __restrict__ bias,
                 const float* __restrict__ addp,
                 float* __restrict__ out, int fo,
                 int kdim, int mtiles, int ntiles, int act) {
  int wave = blockIdx.x * (blockDim.x >> 5) + (threadIdx.x >> 5);
  if (wave >= mtiles * ntiles) return;           // wave-uniform: EXEC stays all-ones
  int tm = wave / ntiles, tn = wave - tm * ntiles;
  int lane = threadIdx.x & 31;
  int m = lane & 15;          // A row within tile / B-D column within tile
  int half = lane >> 4;       // lane-half selects K pair (A/B) or M+8 (C/D)

  const float* Arow = A + (size_t)(tm * 16 + m) * lda;
  const float* Bcol = B + (tn * 16 + m);

  v8f acc = {};
  for (int k0 = 0; k0 < kdim; k0 += 4) {
    int ka = k0 + 2 * half;
    v2f av, bv;
    av.x = Arow[ka];
    av.y = Arow[ka + 1];
    bv.x = Bcol[(size_t)ka * ldb];
    bv.y = Bcol[(size_t)(ka + 1) * ldb];
    acc = __builtin_amdgcn_wmma_f32_16x16x4_f32(
        /*neg_a=*/false, av, /*neg_b=*/false, bv,
        /*c_mod=*/(short)0, acc, /*reuse_a=*/false, /*reuse_b=*/false);
  }

  int col = tn * 16 + m;
  if (col < fo) {
    float bb = bias[col];
#pragma unroll
    for (int r = 0; r < 8; ++r) {
      int row = tm * 16 + r + 8 * half;
      float v = acc[r] + bb;
      if (addp) v += addp[(size_t)row * fo + col];
      if (act) v = fmaxf(v, 0.f);
      out[(size_t)row * fo + col] = v;
    }
  }
}

// ---------------- InstanceNorm (global over N*F) ----------------

__global__ __launch_bounds__(256)
void k_stats(const float* __restrict__ x, int n, float* __restrict__ stats) {
  __shared__ float sh0[256], sh1[256];
  float s = 0.f, q = 0.f;
  for (int i = blockIdx.x * blockDim.x + threadIdx.x; i < n; i += gridDim.x * blockDim.x) {
    float v = x[i];
    s += v; q += v * v;
  }
  sh0[threadIdx.x] = s; sh1[threadIdx.x] = q;
  __syncthreads();
  for (int st = 128; st > 0; st >>= 1) {
    if (threadIdx.x < st) {
      sh0[threadIdx.x] += sh0[threadIdx.x + st];
      sh1[threadIdx.x] += sh1[threadIdx.x + st];
    }
    __syncthreads();
  }
  if (threadIdx.x == 0) {
    fadd_agent(&stats[0], sh0[0]);
    fadd_agent(&stats[1], sh1[0]);
  }
}

__global__ void k_inorm(float* __restrict__ x, int n, const float* __restrict__ stats,
                        const float* __restrict__ gamma, const float* __restrict__ beta,
                        int gi, int relu) {
  int i = blockIdx.x * blockDim.x + threadIdx.x;
  if (i >= n) return;
  float m = stats[0] / (float)n;
  float v = stats[1] / (float)n - m * m;
  float rs = rsqrtf(v + 1e-3f);
  float t = (x[i] - m) * rs * gamma[gi] + beta[gi];
  if (relu) t = fmaxf(t, 0.f);
  x[i] = t;
}

__global__ void k_combine(float* __restrict__ h, const float* __restrict__ h1, int n) {
  int i = blockIdx.x * blockDim.x + threadIdx.x;
  if (i < n) h[i] = fmaxf((h[i] + h1[i]) * 0.5f, 0.f);
}

// ---------------- host-side Cheb layer helper ----------------

static void run_cheb(const float* xin, int fin, int fo,
                     const float* W, const float* bias, const float* addp,
                     float* out, int act,
                     const int* src, const int* dst, const float* wedge,
                     float* T, float* Wpad, hipStream_t s) {
  int finp = (fin + 3) & ~3;        // per-segment K padding (16B aligned segments)
  int kpad = 3 * finp;
  int fpad = (fo + 15) & ~15;
  if (xin) {
    int totT = N_NODES * kpad;
    k_fill_t0<<<cdiv(totT, 256), 256, 0, s>>>(xin, T, fin, kpad, totT);
  }
  int fin4 = finp >> 2;
  int totE = N_EDGES * fin4;
  // t1 = L(t0)
  k_spmm4<<<cdiv(totE, 256), 256, 0, s>>>(src, dst, wedge, T, fin4, kpad, 0, finp, 1.f, totE);
  // t2 = 2*L(t1) ...
  k_spmm4<<<cdiv(totE, 256), 256, 0, s>>>(src, dst, wedge, T, fin4, kpad, finp, 2 * finp, 2.f, totE);
  // ... - t0
  int totF = N_NODES * finp;
  k_t2fix<<<cdiv(totF, 256), 256, 0, s>>>(T, finp, kpad, totF);
  int totW = kpad * fpad;
  k_packw<<<cdiv(totW, 256), 256, 0, s>>>(W, Wpad, fin, finp, fo, fpad, kpad);
  int mtiles = N_NODES / 16, ntiles = fpad / 16;
  int waves = mtiles * ntiles;
  k_gemm_wmma<<<cdiv(waves, 8), 256, 0, s>>>(T, kpad, Wpad, fpad, bias, addp,
                                             out, fo, kpad, mtiles, ntiles, act);
}

// ---------------- entry point ----------------

extern "C" void kernel_launch(void* const* d_in, const int* in_sizes, int n_in,
                              void* d_out, int out_size, void* d_ws, size_t ws_size,
                              hipStream_t stream) {
  const float* gf    = (const float*)d_in[0];   // [N,96]
  const float* enc   = (const float*)d_in[1];   // [N,32]
  const float* prev  = (const float*)d_in[2];   // [N,3]
  const int*   eidx  = (const int*)d_in[3];     // [2,E]
  const float* Wp    = (const float*)d_in[4];   // [3,96,96]
  const float* bp    = (const float*)d_in[5];   // [96]
  const float* Wd0   = (const float*)d_in[6];   // [3,131,64]
  const float* bd0   = (const float*)d_in[7];   // [64]
  const float* Wd    = (const float*)d_in[8];   // [8,3,64,64]
  const float* bd    = (const float*)d_in[9];   // [8,64]
  const float* gamma = (const float*)d_in[10];  // [9]
  const float* beta  = (const float*)d_in[11];  // [9]
  const float* Wout  = (const float*)d_in[12];  // [3,64,3]
  const float* bout  = (const float*)d_in[13];  // [3]

  float* outx = (float*)d_out;                        // [N,64]
  float* outr = outx + (size_t)N_NODES * 64;          // [N,3]

  const int* src = eidx;
  const int* dst = eidx + N_EDGES;

  float* ws = (float*)d_ws;
  size_t o = 0;
  float* deg   = ws + o; o += N_NODES;
  float* wed   = ws + o; o += N_EDGES;
  float* T     = ws + o; o += (size_t)N_NODES * 396;  // [t0|t1|t2] padded, max kpad=396
  float* Wpad  = ws + o; o += (size_t)396 * 96;
  float* H     = ws + o; o += (size_t)N_NODES * 96;
  float* H1    = ws + o; o += (size_t)N_NODES * 64;
  float* stats = ws + o; o += 8;

  // degrees & symmetric-normalized (negated) edge weights
  k_zero<<<cdiv(N_NODES, 256), 256, 0, stream>>>(deg, N_NODES);
  k_deg<<<cdiv(N_EDGES, 256), 256, 0, stream>>>(src, deg, N_EDGES);
  k_wedge<<<cdiv(N_EDGES, 256), 256, 0, stream>>>(src, dst, deg, wed, N_EDGES);

  // process conv: gf -> H [N,96], fused ReLU
  run_cheb(gf, 96, 96, Wp, bp, nullptr, H, 1, src, dst, wed, T, Wpad, stream);

  // 3 residual deform blocks x 3 convs
  const int nH = N_NODES * 64;
  for (int blk = 0; blk < 3; ++blk) {
    for (int i = 0; i < 3; ++i) {
      int li = blk * 3 + i;
      if (li == 0) {
        int kpad = 396, tot = N_NODES * kpad;  // finp = 132
        k_fill_t0_cat<<<cdiv(tot, 256), 256, 0, stream>>>(H, enc, prev, T, kpad, tot);
        run_cheb(nullptr, 131, 64, Wd0, bd0, nullptr, H, 0, src, dst, wed, T, Wpad, stream);
      } else {
        run_cheb(H, 64, 64, Wd + (size_t)(li - 1) * 3 * 64 * 64, bd + (size_t)(li - 1) * 64,
                 nullptr, H, 0, src, dst, wed, T, Wpad, stream);
      }
      // global InstanceNorm (+ fused ReLU only for i==1, matching reference)
      k_zero<<<1, 32, 0, stream>>>(stats, 2);
      k_stats<<<512, 256, 0, stream>>>(H, nH, stats);
      k_inorm<<<cdiv(nH, 256), 256, 0, stream>>>(H, nH, stats, gamma, beta, li, (i == 1) ? 1 : 0);
      if (i == 0) k_copy<<<cdiv(nH, 256), 256, 0, stream>>>(H, H1, nH);
    }
    k_combine<<<cdiv(nH, 256), 256, 0, stream>>>(H, H1, nH);
  }

  // first output: x [N,64]
  k_copy<<<cdiv(nH, 256), 256, 0, stream>>>(H, outx, nH);
  // out conv: H -> res [N,3], residual add of prev_results
  run_cheb(H, 64, 3, Wout, bout, prev, outr, 0, src, dst, wed, T, Wpad, stream);
}